// ContrastiveLoss_38912403702379
// MI455X (gfx1250) — compile-verified
//
#include <hip/hip_runtime.h>

typedef __attribute__((ext_vector_type(2))) float v2f;
typedef __attribute__((ext_vector_type(8))) float v8f;

#define NCLS   1000
#define HW     196
#define HALF   98
#define NBATCH 64
#define LOGEPS -13.815510557964274f   // log(1e-6)

// ---------------------------------------------------------------------------
// Stage 1: per-(b, 16-class tile) row statistics.
// Block = 128 threads = 4 waves; wave w handles classes [ (grp*4+w)*16, +16 ).
// Lane l: half h = l>>4 (98 contiguous pixels, register-resident) of row
// r = l&15.  Outputs per (b,c): ablation[b,c] and
// S1[b,c] = sum over fg pixels of clipped log softmax(4*cc).
// ---------------------------------------------------------------------------
__global__ __launch_bounds__(128) void cl_stage1(
    const float* __restrict__ cams,
    const int*   __restrict__ y0,
    const int*   __restrict__ labels,
    float* __restrict__ abl_out,
    float* __restrict__ s1_out)
{
  __shared__ __align__(16) float sT[HW];   // target-class CAM row for this b
  __shared__ __align__(16) float sF[HW];   // fg mask as float

  const int tid  = threadIdx.x;
  const int wave = tid >> 5;
  const int lane = tid & 31;
  const int b    = blockIdx.x >> 4;        // 16 tile-groups per batch
  const int grp  = blockIdx.x & 15;
  const int c0   = (grp * 4 + wave) * 16;
  const int r    = lane & 15;
  const int h    = lane >> 4;

  const int lab = labels[b];
  for (int e = tid; e < HW; e += 128) {
    sT[e] = cams[((size_t)b * NCLS + lab) * HW + e];
    sF[e] = (float)y0[b * HW + e];
  }
  __syncthreads();

  int c = c0 + r;
  if (c > NCLS - 1) c = NCLS - 1;          // clamp; invalid rows never store

  const float2* row2 = (const float2*)(cams + ((size_t)b * NCLS + c) * HW + h * HALF);
  const float2* tp2  = (const float2*)(sT + h * HALF);
  const float2* fp2  = (const float2*)(sF + h * HALF);

  // Pass A: cc (registers), ablation partial, row-max partial, fg bitmask
  float    cc[HALF];
  unsigned fgm[4] = {0u, 0u, 0u, 0u};
  float    abl = 0.0f;
  float    m   = -3.0e38f;
  #pragma unroll
  for (int i = 0; i < HALF / 2; ++i) {
    float2 v = row2[i];
    float2 t = tp2[i];
    float2 f = fp2[i];
    float c0v = v.x - t.x;
    float c1v = v.y - t.y;
    cc[2 * i]     = c0v;
    cc[2 * i + 1] = c1v;
    abl += (f.x != 0.0f) ? -c0v : fabsf(c0v);
    abl += (f.y != 0.0f) ? -c1v : fabsf(c1v);
    m = fmaxf(m, fmaxf(c0v, c1v));
    fgm[(2 * i) >> 5]     |= ((f.x != 0.0f) ? 1u : 0u) << ((2 * i) & 31);
    fgm[(2 * i + 1) >> 5] |= ((f.y != 0.0f) ? 1u : 0u) << ((2 * i + 1) & 31);
  }
  m = fmaxf(m, __shfl_xor(m, 16, 32));     // full-row max

  // Pass B: softmax normalizer of 4*cc; cache 4*(cc-m) back into cc[]
  float z = 0.0f;
  #pragma unroll
  for (int j = 0; j < HALF; ++j) {
    float t4 = 4.0f * (cc[j] - m);
    cc[j] = t4;
    z += __expf(t4);
  }
  z += __shfl_xor(z, 16, 32);
  float lz = __logf(z);

  // Pass C: masked clipped-log sum
  float s1 = 0.0f;
  #pragma unroll
  for (int j = 0; j < HALF; ++j) {
    float lp = fmaxf(cc[j] - lz, LOGEPS);
    s1 += ((fgm[j >> 5] >> (j & 31)) & 1u) ? lp : 0.0f;
  }

  // Cross-half 16-row reductions via WMMA: D[m,n] = sum_k A[m,k] * 1.
  // A 16x4 f32 layout: lanes 0-15 supply K0/K1, lanes 16-31 supply K2/K3,
  // M = lane % 16 -> one WMMA sums both halves of all 16 rows.
  v2f a_abl; a_abl[0] = abl; a_abl[1] = 0.0f;
  v2f a_s1;  a_s1[0]  = s1;  a_s1[1]  = 0.0f;
  v2f ones;  ones[0]  = 1.0f; ones[1] = 1.0f;
  v8f zacc = {};
  v8f d_abl = __builtin_amdgcn_wmma_f32_16x16x4_f32(
      false, a_abl, false, ones, (short)0, zacc, false, false);
  v8f d_s1  = __builtin_amdgcn_wmma_f32_16x16x4_f32(
      false, a_s1,  false, ones, (short)0, zacc, false, false);

  // C/D layout: VGPR j = row j (lanes 0-15) / row j+8 (lanes 16-31),
  // replicated over all N -> lane 0 stores rows 0-7, lane 16 stores rows 8-15.
  if (lane == 0 || lane == 16) {
    const int rbase = c0 + (lane ? 8 : 0);
    #pragma unroll
    for (int j = 0; j < 8; ++j) {
      int cc2 = rbase + j;
      if (cc2 < NCLS) {
        abl_out[b * NCLS + cc2] = d_abl[j];
        s1_out [b * NCLS + cc2] = d_s1[j];
      }
    }
  }
}

// ---------------------------------------------------------------------------
// Stage 2: per-b class reductions -> ace_b, kld_b
// ---------------------------------------------------------------------------
__device__ __forceinline__ float block_sum(float v, float* red, int tid) {
  red[tid] = v; __syncthreads();
  #pragma unroll
  for (int s = 128; s > 0; s >>= 1) {
    if (tid < s) red[tid] += red[tid + s];
    __syncthreads();
  }
  float r = red[0];
  __syncthreads();
  return r;
}

__device__ __forceinline__ float block_max(float v, float* red, int tid) {
  red[tid] = v; __syncthreads();
  #pragma unroll
  for (int s = 128; s > 0; s >>= 1) {
    if (tid < s) red[tid] = fmaxf(red[tid], red[tid + s]);
    __syncthreads();
  }
  float r = red[0];
  __syncthreads();
  return r;
}

__global__ __launch_bounds__(256) void cl_stage2(
    const float* __restrict__ abl,
    const float* __restrict__ s1,
    const int*   __restrict__ y0,
    const int*   __restrict__ labels,
    float* __restrict__ ace_out,
    float* __restrict__ kld_out)
{
  __shared__ float red[256];
  __shared__ float sh_lab;

  const int b   = blockIdx.x;
  const int tid = threadIdx.x;
  const int lab = labels[b];

  float aval[4];
  float lmax = -3.0e38f, lsa = 0.0f, lss = 0.0f;
  #pragma unroll
  for (int k = 0; k < 4; ++k) {
    int c = tid + (k << 8);
    bool v = (c < NCLS);
    float av = v ? abl[b * NCLS + c] : -3.0e38f;
    float sv = v ? s1 [b * NCLS + c] : 0.0f;
    aval[k] = av;
    lmax = fmaxf(lmax, av);
    lsa += v ? av : 0.0f;
    lss += sv;
    if (v && c == lab) sh_lab = av;
  }
  float nf_part = (tid < HW) ? (float)y0[b * HW + tid] : 0.0f;

  float M = block_max(lmax, red, tid);

  float lz = 0.0f;
  #pragma unroll
  for (int k = 0; k < 4; ++k) lz += __expf(aval[k] - M);

  float Z       = block_sum(lz,  red, tid);
  float sum_abl = block_sum(lsa, red, tid);
  float sum_s1  = block_sum(lss, red, tid);
  float nfg     = block_sum(nf_part, red, tid);

  if (tid == 0) {
    float lse = M + __logf(Z);
    float abl_lab = sh_lab;
    // smoothed CE: -(0.9*logp[lab] + (0.1/C)*sum_c logp[c])
    float ace = -(0.9f * (abl_lab - lse) +
                  (0.1f / (float)NCLS) * (sum_abl - (float)NCLS * lse));
    // fg_probs value at fg pixels (class-independent)
    float fgp = 1.0f / (nfg + ((float)HW - nfg) * __expf(-10.0f));
    float kld = -fgp * sum_s1;
    ace_out[b] = ace;
    kld_out[b] = kld;
  }
}

// ---------------------------------------------------------------------------
// Stage 3: final scalar, deterministic single-block combine
// ---------------------------------------------------------------------------
__global__ __launch_bounds__(64) void cl_stage3(
    const float* __restrict__ ace,
    const float* __restrict__ kld,
    float* __restrict__ out)
{
  __shared__ float sa[64];
  __shared__ float sk[64];
  const int t = threadIdx.x;
  sa[t] = ace[t];
  sk[t] = kld[t];
  __syncthreads();
  #pragma unroll
  for (int s = 32; s > 0; s >>= 1) {
    if (t < s) { sa[t] += sa[t + s]; sk[t] += sk[t + s]; }
    __syncthreads();
  }
  if (t == 0)
    out[0] = 1.0f * (sa[0] / (float)NBATCH) + 0.5f * (sk[0] / (float)NBATCH);
}

// ---------------------------------------------------------------------------
extern "C" void kernel_launch(void* const* d_in, const int* in_sizes, int n_in,
                              void* d_out, int out_size, void* d_ws, size_t ws_size,
                              hipStream_t stream) {
  const float* cams   = (const float*)d_in[0];  // [64,1000,14,14]
  const int*   y0     = (const int*)  d_in[1];  // [64,14,14]
  const int*   labels = (const int*)  d_in[2];  // [64]
  float* out = (float*)d_out;

  char* ws = (char*)d_ws;
  float* abl = (float*)(ws);
  float* s1  = (float*)(ws + (size_t)NBATCH * NCLS * sizeof(float));
  float* ace = (float*)(ws + (size_t)2 * NBATCH * NCLS * sizeof(float));
  float* kld = (float*)(ws + (size_t)2 * NBATCH * NCLS * sizeof(float)
                           + NBATCH * sizeof(float));

  cl_stage1<<<dim3(NBATCH * 16), dim3(128), 0, stream>>>(cams, y0, labels, abl, s1);
  cl_stage2<<<dim3(NBATCH), dim3(256), 0, stream>>>(abl, s1, y0, labels, ace, kld);
  cl_stage3<<<dim3(1), dim3(64), 0, stream>>>(ace, kld, out);
}